// ASSC_66657892434080
// MI455X (gfx1250) — compile-verified
//
#include <hip/hip_runtime.h>
#include <stdint.h>

typedef __attribute__((ext_vector_type(2))) float v2f;
typedef __attribute__((ext_vector_type(8))) float v8f;

#define kB   8
#define kC   256
#define kC8  32
#define kH   96
#define kHW  (96 * 96)
#define kP   280
#define kPP  288

#if __has_builtin(__builtin_amdgcn_global_load_async_to_lds_b128) && \
    __has_builtin(__builtin_amdgcn_s_wait_asynccnt)
#define ASSC_ASYNC 1
#else
#define ASSC_ASYNC 0
#endif

static __device__ __forceinline__ v8f wmma4(v2f a, v2f b, v8f c) {
  // D = A(16x4 f32) * B(4x16 f32) + C(16x16 f32)
  return __builtin_amdgcn_wmma_f32_16x16x4_f32(false, a, false, b, (short)0, c,
                                               false, false);
}

// ---------------- PSP adaptive pooling: [B,C,H,W] -> [B,C,288] (280 valid) --------
__global__ void assc_psp(const float* __restrict__ x1, const float* __restrict__ x,
                         float* __restrict__ Pp1, float* __restrict__ Pp2) {
  int id = blockIdx.x;
  const float* src;
  float* dst;
  if (id < kB * kC) { src = x1; dst = Pp1; }
  else              { src = x;  dst = Pp2; id -= kB * kC; }
  const float* tile = src + (size_t)id * kHW;

  __shared__ float sm[kHW];      // 36 KB image tile
  __shared__ float rs[30 * kH];  // row-pooled sums for all 1+3+5+7+14 = 30 bins
  int t = threadIdx.x;
#if ASSC_ASYNC
  {
    // gfx1250 async copy engine: global -> LDS without VGPR staging (ASYNCcnt).
    // Builtin signature (from hipcc diagnostic): param0 = AS1 v4i*, param1 = AS3 v4i*,
    // then imm offset + imm cpol.
    typedef int v4i_vs __attribute__((vector_size(16)));
    typedef __attribute__((address_space(1))) v4i_vs* gvec_p;
    typedef __attribute__((address_space(3))) v4i_vs* lvec_p;
    gvec_p g = (gvec_p)tile;
    lvec_p l = (lvec_p)sm;
    for (int i = t; i < (kHW * 4) / 16; i += 256) {
      __builtin_amdgcn_global_load_async_to_lds_b128(g + i, l + i, 0, 0);
    }
    __builtin_amdgcn_s_wait_asynccnt(0);  // our wave's LDS writes are done
  }
#else
  for (int i = t; i < kHW; i += 256) sm[i] = tile[i];
#endif
  __syncthreads();

  // row bins (exact torch adaptive bin boundaries)
  for (int task = t; task < 30 * kH; task += 256) {
    int r = task / kH, w = task % kH;
    int s, i;
    if (r < 1)       { s = 1;  i = r;      }
    else if (r < 4)  { s = 3;  i = r - 1;  }
    else if (r < 9)  { s = 5;  i = r - 4;  }
    else if (r < 16) { s = 7;  i = r - 9;  }
    else             { s = 14; i = r - 16; }
    int hs = (i * kH) / s, he = ((i + 1) * kH + s - 1) / s;
    float acc = 0.f;
    for (int h = hs; h < he; ++h) acc += sm[h * kH + w];
    rs[r * kH + w] = acc;
  }
  __syncthreads();

  // column bins -> 280 pooled averages (+ zero pad to 288)
  for (int p = t; p < kPP; p += 256) {
    float val = 0.f;
    if (p < kP) {
      int s, pof, rbase;
      if (p < 1)       { s = 1;  pof = 0;  rbase = 0;  }
      else if (p < 10) { s = 3;  pof = 1;  rbase = 1;  }
      else if (p < 35) { s = 5;  pof = 10; rbase = 4;  }
      else if (p < 84) { s = 7;  pof = 35; rbase = 9;  }
      else             { s = 14; pof = 84; rbase = 16; }
      int rel = p - pof, i = rel / s, j = rel % s;
      int hs = (i * kH) / s, he = ((i + 1) * kH + s - 1) / s;
      int wss = (j * kH) / s, we = ((j + 1) * kH + s - 1) / s;
      const float* rrow = &rs[(rbase + i) * kH];
      float acc = 0.f;
      for (int w = wss; w < we; ++w) acc += rrow[w];
      val = acc / (float)((he - hs) * (we - wss));
    }
    dst[(size_t)id * kPP + p] = val;
  }
}

// ------------- con[j,c,kh,kw] -> conR[tap][j][c] (contiguous K for A loads) -------
__global__ void assc_reorg(const float* __restrict__ con, float* __restrict__ conR) {
  int i = blockIdx.x * 256 + threadIdx.x;
  if (i >= 9 * kC8 * kC) return;
  int c = i % kC;
  int j = (i / kC) % kC8;
  int tap = i / (kC * kC8);
  conR[i] = con[((size_t)j * kC + c) * 9 + tap];
}

// --------- D[b] = W[M,K] @ S[b][K,288] + bias (pad cols forced to 0) -------------
__global__ void assc_proj_gemm(const float* __restrict__ W, const float* __restrict__ bias,
                               const float* __restrict__ S, float* __restrict__ D,
                               int M, int K, int Nvalid) {
  int wid = (blockIdx.x * blockDim.x + threadIdx.x) >> 5;
  int lane = threadIdx.x & 31;
  int mtiles = M >> 4;
  int nt = wid % (kPP / 16);
  int mt = (wid / (kPP / 16)) % mtiles;
  int b = wid / ((kPP / 16) * mtiles);
  int row = lane & 15;
  int kb = (lane >> 4) << 1;  // K sub-offset 0 or 2 (A/B f32 layout)

  const float* Wm = W + (size_t)(mt * 16 + row) * K;
  v8f acc = {};
  for (int k0 = 0; k0 < K; k0 += 4) {
    v2f a;  a.x = Wm[k0 + kb];            a.y = Wm[k0 + kb + 1];
    const float* sp = S + ((size_t)b * K + (k0 + kb)) * kPP + nt * 16 + row;
    v2f bb; bb.x = sp[0];                 bb.y = sp[kPP];
    acc = wmma4(a, bb, acc);
  }
  int col = lane & 15;
  int rbase = (lane >> 4) << 3;
#pragma unroll
  for (int r = 0; r < 8; ++r) {
    int m = mt * 16 + rbase + r;
    int n = nt * 16 + col;
    float v = (n < Nvalid) ? (acc[r] + bias[m]) : 0.f;  // keep pads exactly zero
    D[((size_t)b * M + m) * kPP + n] = v;
  }
}

// --------- aff[b][m][j] = sigmoid( PQ[b][m][:] . PK[b][j][:] ), K=288 ------------
__global__ void assc_aff(const float* __restrict__ PQ, const float* __restrict__ PK,
                         float* __restrict__ aff) {
  int wid = (blockIdx.x * blockDim.x + threadIdx.x) >> 5;
  int lane = threadIdx.x & 31;
  int nt = wid & 1;
  int mt = (wid >> 1) & 15;
  int b = wid >> 5;
  int row = lane & 15;
  int kb = (lane >> 4) << 1;

  const float* Aq = PQ + ((size_t)b * kC + mt * 16 + row) * kPP;
  const float* Bk = PK + ((size_t)b * kC8 + nt * 16 + row) * kPP;  // B = PK^T
  v8f acc = {};
  for (int k0 = 0; k0 < kPP; k0 += 4) {
    v2f a;  a.x = Aq[k0 + kb];  a.y = Aq[k0 + kb + 1];
    v2f bb; bb.x = Bk[k0 + kb]; bb.y = Bk[k0 + kb + 1];
    acc = wmma4(a, bb, acc);
  }
  int col = lane & 15;
  int rbase = (lane >> 4) << 3;
#pragma unroll
  for (int r = 0; r < 8; ++r) {
    int m = mt * 16 + rbase + r;
    aff[((size_t)b * kC + m) * kC8 + nt * 16 + col] =
        1.0f / (1.0f + __expf(-acc[r]));
  }
}

// --------- y[b][32][9216] = conv3x3(x_1[b], con) via implicit GEMM ---------------
// One wave -> 32x32 output tile: 2 M-tiles x 2 N-tiles; A regs shared across N,
// B regs shared across M -> 4 WMMAs per 8 loaded dwords per lane.
__global__ void assc_conv(const float* __restrict__ x1, const float* __restrict__ conR,
                          float* __restrict__ y) {
  int wid = (blockIdx.x * blockDim.x + threadIdx.x) >> 5;
  int lane = threadIdx.x & 31;
  int np = wid % (kHW / 32);   // 288 column-pair tiles
  int b = wid / (kHW / 32);
  int h = np / 3;              // 96/32 == 3 pair-tiles per image row
  int w0 = (np % 3) * 32;
  int row = lane & 15;
  int kb = (lane >> 4) << 1;
  int col = lane & 15;

  v8f acc00 = {}, acc01 = {}, acc10 = {}, acc11 = {};
  const float* xb0 = x1 + (size_t)b * kC * kHW;
  for (int tap = 0; tap < 9; ++tap) {
    int dh = tap / 3 - 1, dw = tap % 3 - 1;
    int hh = h + dh;
    bool rok = (hh >= 0) && (hh < kH);
    int wsh0 = w0 + col + dw;
    int wsh1 = wsh0 + 16;
    bool ok0 = rok && (wsh0 >= 0) && (wsh0 < kH);
    bool ok1 = rok && (wsh1 >= 0) && (wsh1 < kH);
    const float* A0 = conR + (size_t)(tap * kC8 + row) * kC;
    const float* A1 = A0 + 16 * kC;
    const float* xr = xb0 + hh * kH;  // only dereferenced when ok
    for (int k0 = 0; k0 < kC; k0 += 4) {
      v2f a0; a0.x = A0[k0 + kb]; a0.y = A0[k0 + kb + 1];
      v2f a1; a1.x = A1[k0 + kb]; a1.y = A1[k0 + kb + 1];
      const float* xp = xr + (size_t)(k0 + kb) * kHW;
      v2f b0, b1;
      b0.x = ok0 ? xp[wsh0] : 0.f;
      b0.y = ok0 ? xp[wsh0 + kHW] : 0.f;
      b1.x = ok1 ? xp[wsh1] : 0.f;
      b1.y = ok1 ? xp[wsh1 + kHW] : 0.f;
      acc00 = wmma4(a0, b0, acc00);
      acc10 = wmma4(a1, b0, acc10);
      acc01 = wmma4(a0, b1, acc01);
      acc11 = wmma4(a1, b1, acc11);
    }
  }
  int rbase = (lane >> 4) << 3;
  size_t n0 = (size_t)np * 32 + col;
#pragma unroll
  for (int r = 0; r < 8; ++r) {
    y[((size_t)b * kC8 + rbase + r) * kHW + n0] = acc00[r];
    y[((size_t)b * kC8 + rbase + r) * kHW + n0 + 16] = acc01[r];
    y[((size_t)b * kC8 + 16 + rbase + r) * kHW + n0] = acc10[r];
    y[((size_t)b * kC8 + 16 + rbase + r) * kHW + n0 + 16] = acc11[r];
  }
}

// --------- out[b][256][9216] = aff[b][256,32] @ y[b][32,9216] --------------------
// One wave -> 128x16 output tile (8 M-tiles share each B load of y -> 8x less
// y re-read traffic, 8 WMMAs per B dword-pair).
__global__ void assc_mix(const float* __restrict__ aff, const float* __restrict__ y,
                         float* __restrict__ out) {
  int wid = (blockIdx.x * blockDim.x + threadIdx.x) >> 5;
  int lane = threadIdx.x & 31;
  int nt = wid % (kHW / 16);
  int mg = (wid / (kHW / 16)) & 1;  // M half: rows [mg*128, mg*128+128)
  int b = wid / ((kHW / 16) * 2);
  int row = lane & 15;
  int col = lane & 15;
  int kb = (lane >> 4) << 1;

  const float* Ab = aff + ((size_t)b * kC + mg * 128 + row) * kC8;
  const float* Yb = y + (size_t)b * kC8 * kHW + nt * 16 + col;
  v8f acc[8] = {};
#pragma unroll
  for (int k0 = 0; k0 < kC8; k0 += 4) {
    v2f bb;
    bb.x = Yb[(size_t)(k0 + kb) * kHW];
    bb.y = Yb[(size_t)(k0 + kb + 1) * kHW];
#pragma unroll
    for (int m = 0; m < 8; ++m) {
      v2f a;
      a.x = Ab[(size_t)m * 16 * kC8 + k0 + kb];
      a.y = Ab[(size_t)m * 16 * kC8 + k0 + kb + 1];
      acc[m] = wmma4(a, bb, acc[m]);
    }
  }
  int rbase = (lane >> 4) << 3;
#pragma unroll
  for (int m = 0; m < 8; ++m) {
#pragma unroll
    for (int r = 0; r < 8; ++r) {
      out[((size_t)b * kC + mg * 128 + m * 16 + rbase + r) * kHW + nt * 16 + col] =
          acc[m][r];
    }
  }
}

extern "C" void kernel_launch(void* const* d_in, const int* in_sizes, int n_in,
                              void* d_out, int out_size, void* d_ws, size_t ws_size,
                              hipStream_t stream) {
  const float* x1 = (const float*)d_in[0];
  const float* x = (const float*)d_in[1];
  const float* wq = (const float*)d_in[2];
  const float* bq = (const float*)d_in[3];
  const float* wk = (const float*)d_in[4];
  const float* bk = (const float*)d_in[5];
  const float* con = (const float*)d_in[6];
  float* out = (float*)d_out;

  // workspace layout (floats): ~17.4 MB total
  float* ws = (float*)d_ws;
  float* Pp1 = ws;                            // [8][256][288]
  float* Pp2 = Pp1 + (size_t)kB * kC * kPP;   // [8][256][288]
  float* PQ  = Pp2 + (size_t)kB * kC * kPP;   // [8][256][288]
  float* PK  = PQ + (size_t)kB * kC * kPP;    // [8][32][288]
  float* aff = PK + (size_t)kB * kC8 * kPP;   // [8][256][32]
  float* cnR = aff + (size_t)kB * kC * kC8;   // [9][32][256]
  float* y   = cnR + (size_t)9 * kC8 * kC;    // [8][32][9216]

  assc_psp<<<2 * kB * kC, 256, 0, stream>>>(x1, x, Pp1, Pp2);
  assc_reorg<<<(9 * kC8 * kC + 255) / 256, 256, 0, stream>>>(con, cnR);
  // PQ = wq @ PSP(x_1) + bq   (pooling commutes with the 1x1 conv)
  assc_proj_gemm<<<(kB * (kC / 16) * (kPP / 16)) / 4, 128, 0, stream>>>(
      wq, bq, Pp1, PQ, kC, kC, kP);
  // PK = wk @ PSP(x) + bk
  assc_proj_gemm<<<(kB * (kC8 / 16) * (kPP / 16)) / 4, 128, 0, stream>>>(
      wk, bk, Pp2, PK, kC8, kC, kP);
  assc_aff<<<(kB * (kC / 16) * (kC8 / 16)) / 4, 128, 0, stream>>>(PQ, PK, aff);
  assc_conv<<<(kB * (kHW / 32)) / 4, 128, 0, stream>>>(x1, cnR, y);
  assc_mix<<<(kB * 2 * (kHW / 16)) / 4, 128, 0, stream>>>(aff, y, out);
}